// GINConv_8057358647608
// MI455X (gfx1250) — compile-verified
//
#include <hip/hip_runtime.h>

typedef __attribute__((ext_vector_type(2))) float v2f;
typedef __attribute__((ext_vector_type(8))) float v8f;

#define N_NODES 50000
#define N_EDGES 800000
#define D_IN    64
#define D_HID   128
#define HPAD    132   // padded LDS row stride (floats) to avoid bank conflicts

// ---------------------------------------------------------------------------
// Kernel 1: agg[i] = (1 + eps) * x[i]   (fuses the GIN self-term into the
// accumulator so the scatter kernel can atomically add straight into it)
// ---------------------------------------------------------------------------
__global__ void gin_init(const float* __restrict__ x,
                         const float* __restrict__ eps,
                         float* __restrict__ agg, int n) {
    int i = blockIdx.x * blockDim.x + threadIdx.x;
    if (i < n) agg[i] = (1.0f + eps[0]) * x[i];
}

// ---------------------------------------------------------------------------
// Kernel 2: for each edge e: agg[row[e]] += x[col[e]]
// One thread per (edge, 16-byte feature chunk): 16 lanes cover one edge's
// 256B feature row -> coalesced float4 gather, then 4 f32 global atomics
// (L2-resident: x and agg are 12.8 MB each, far under the 192 MB L2).
// ---------------------------------------------------------------------------
__global__ void gin_scatter(const float* __restrict__ x,
                            const int* __restrict__ ei,   // [2, E] flattened
                            float* __restrict__ agg) {
    unsigned tid = blockIdx.x * blockDim.x + threadIdx.x;  // [0, E*16)
    unsigned e = tid >> 4;
    if (e >= N_EDGES) return;
    int f = (tid & 15u) << 2;            // feature base (multiple of 4 floats)
    int dst = ei[e];                     // row (destination node)
    int src = ei[N_EDGES + e];           // col (source node)
    const float4 v = *(const float4*)(x + (size_t)src * D_IN + f);
    float* a = agg + (size_t)dst * D_IN + f;
    atomicAdd(a + 0, v.x);
    atomicAdd(a + 1, v.y);
    atomicAdd(a + 2, v.z);
    atomicAdd(a + 3, v.w);
}

// ---------------------------------------------------------------------------
// Kernel 3: out = relu(h @ W1 + b1) @ W2 + b2  with V_WMMA_F32_16X16X4_F32.
// One wave = one 16-row tile (50000 = 3125 full tiles -> EXEC all-ones in
// every WMMA wave). 4 waves / block, 33 KB LDS for the transposed hidden tile.
// ---------------------------------------------------------------------------
__global__ __launch_bounds__(128)
void gin_mlp(const float* __restrict__ h,    // [N, 64]  (agg after scatter)
             const float* __restrict__ W1,   // [64, 128]
             const float* __restrict__ b1,   // [128]
             const float* __restrict__ W2,   // [128, 64]
             const float* __restrict__ b2,   // [64]
             float* __restrict__ out) {      // [N, 64]
    __shared__ float hid[4][16 * HPAD];

    const int wave  = threadIdx.x >> 5;
    const int lane  = threadIdx.x & 31;
    const int r     = lane & 15;            // row (A) / column (B,D) index
    const int koff  = (lane >> 4) << 1;     // 0 for lanes 0-15, 2 for 16-31
    const int mhalf = (lane >> 4) << 3;     // D rows: +0 or +8
    const int tile  = blockIdx.x * 4 + wave;
    if (tile * 16 >= N_NODES) return;       // wave-uniform exit
    const int m0 = tile * 16;

    // ---- Layer 1: hid = relu(h[m0:m0+16] @ W1 + b1), 16 x 128 ----
    // Preload all A fragments for K=64 (16 k-steps of 4): lane<16 holds
    // (row r, K=k..k+1), lane>=16 holds (row r, K=k+2..k+3).
    v2f afr[16];
    const float* arow = h + (size_t)(m0 + r) * D_IN + koff;
#pragma unroll
    for (int kk = 0; kk < 16; ++kk)
        afr[kk] = *(const v2f*)(arow + kk * 4);

    float* hrow = &hid[wave][0];
#pragma unroll
    for (int nt = 0; nt < 8; ++nt) {
        const int n0 = nt * 16;
        const float bias = b1[n0 + r];
        v8f acc;
#pragma unroll
        for (int v = 0; v < 8; ++v) acc[v] = bias;
#pragma unroll
        for (int kk = 0; kk < 16; ++kk) {
            const int k = kk * 4 + koff;
            v2f bfr;                         // B(4x16): lane holds W1[k..k+1][n0+r]
            bfr.x = W1[(size_t)k * D_HID + n0 + r];
            bfr.y = W1[(size_t)(k + 1) * D_HID + n0 + r];
            acc = __builtin_amdgcn_wmma_f32_16x16x4_f32(
                false, afr[kk], false, bfr, (short)0, acc, false, false);
        }
        // D layout: VGPR v -> (M = v + mhalf, N = n0 + r); ReLU into LDS
#pragma unroll
        for (int v = 0; v < 8; ++v) {
            float t = acc[v];
            hrow[(v + mhalf) * HPAD + n0 + r] = t > 0.0f ? t : 0.0f;
        }
    }
    // hid tile is private to this wave; LDS ops are in-order per wave and the
    // compiler inserts the s_wait_dscnt before the dependent reads below.

    // ---- Layer 2: out[m0:m0+16] = hid @ W2 + b2, 16 x 64 ----
    float* orow = out + (size_t)m0 * D_IN;
#pragma unroll
    for (int nt = 0; nt < 4; ++nt) {
        const int n0 = nt * 16;
        const float bias = b2[n0 + r];
        v8f acc;
#pragma unroll
        for (int v = 0; v < 8; ++v) acc[v] = bias;
#pragma unroll
        for (int kk = 0; kk < 32; ++kk) {
            const int k = kk * 4 + koff;
            v2f a2;                          // A frag from transposed LDS tile
            a2.x = hrow[r * HPAD + k];
            a2.y = hrow[r * HPAD + k + 1];
            v2f bfr;
            bfr.x = W2[(size_t)k * D_IN + n0 + r];
            bfr.y = W2[(size_t)(k + 1) * D_IN + n0 + r];
            acc = __builtin_amdgcn_wmma_f32_16x16x4_f32(
                false, a2, false, bfr, (short)0, acc, false, false);
        }
#pragma unroll
        for (int v = 0; v < 8; ++v)
            orow[(size_t)(v + mhalf) * D_IN + n0 + r] = acc[v];
    }
}

// ---------------------------------------------------------------------------
extern "C" void kernel_launch(void* const* d_in, const int* in_sizes, int n_in,
                              void* d_out, int out_size, void* d_ws, size_t ws_size,
                              hipStream_t stream) {
    const float* x   = (const float*)d_in[0];
    const int*   ei  = (const int*)d_in[1];   // [2, 800000] int32 (canonicalized)
    const float* W1  = (const float*)d_in[2];
    const float* b1  = (const float*)d_in[3];
    const float* W2  = (const float*)d_in[4];
    const float* b2  = (const float*)d_in[5];
    const float* eps = (const float*)d_in[6];
    float* out = (float*)d_out;
    float* agg = (float*)d_ws;                // N*64*4 = 12.8 MB scratch

    // 1) agg = (1+eps) * x
    int n = N_NODES * D_IN;
    gin_init<<<(n + 255) / 256, 256, 0, stream>>>(x, eps, agg, n);

    // 2) scatter-add neighbor features (E*16 chunk-threads)
    unsigned total = (unsigned)N_EDGES * 16u;
    gin_scatter<<<(total + 255) / 256, 256, 0, stream>>>(x, ei, agg);

    // 3) fused 2-layer MLP with fp32 WMMA (3125 tiles, 4 waves/block)
    int tiles = (N_NODES + 15) / 16;          // 3125 (all full)
    gin_mlp<<<(tiles + 3) / 4, 128, 0, stream>>>(agg, W1, b1, W2, b2, out);
}